// MyAttentionHead_48163763257486
// MI455X (gfx1250) — compile-verified
//
#include <hip/hip_runtime.h>
#include <math.h>

// Problem constants (match reference: B=8, T=4096, C=1024, H=64)
constexpr int B_ = 8;
constexpr int T_ = 4096;
constexpr int C_ = 1024;
constexpr int H_ = 64;
constexpr int ROW_TILES = (B_ * T_) / 16;   // 2048
constexpr float SCALE = 0.03125f;           // 1/sqrt(C) = 1/32

typedef __attribute__((ext_vector_type(2))) float v2f;
typedef __attribute__((ext_vector_type(8))) float v8f;

// ---------------------------------------------------------------------------
// Kernel 1: fused QKV projection.
// One wave (32 threads) per 16-row tile of x. Reads the x tile once and keeps
// 12 WMMA f32 accumulators (q/k/v  x  4 column tiles of 16) live in VGPRs.
// x (134 MB) is the only true HBM stream; W* (768 KB) are L2-resident.
// ---------------------------------------------------------------------------
__global__ __launch_bounds__(32) void qkv_proj_kernel(
    const float* __restrict__ x,
    const float* __restrict__ Wq,
    const float* __restrict__ Wk,
    const float* __restrict__ Wv,
    float* __restrict__ qo,
    float* __restrict__ ko,
    float* __restrict__ vo)
{
    const int lane = threadIdx.x;        // 0..31
    const int hl   = lane >> 4;          // half: 0 or 1
    const int mn   = lane & 15;          // A: row m ; B/C: col n

    const int tile = blockIdx.x;         // 0..ROW_TILES-1 over B*T
    const long xBase   = (long)tile * 16 * C_;   // first element of row tile
    const long outBase = (long)tile * 16 * H_;

    // A-fragment source pointer for this lane: row mn, k offset 2*hl
    const float* xA = x + xBase + (long)mn * C_ + 2 * hl;

    const float* W[3] = { Wq, Wk, Wv };
    float* outp[3]    = { qo, ko, vo };

    v8f acc[3][4] = {};

    for (int kk = 0; kk < C_; kk += 4) {
        // prefetch the x fragment a few K-steps ahead into the near cache
        // (locality 3 -> WGP-scope global_prefetch_b8)
        if (kk + 16 < C_) __builtin_prefetch(xA + kk + 16, 0, 3);

        v2f a = *(const v2f*)(xA + kk);          // contiguous -> b64 load
        const int krow = kk + 2 * hl;
#pragma unroll
        for (int m = 0; m < 3; ++m) {
            const float* Wm = W[m];
#pragma unroll
            for (int j = 0; j < 4; ++j) {
                v2f bf;
                bf[0] = Wm[(long)(krow    ) * H_ + 16 * j + mn];
                bf[1] = Wm[(long)(krow + 1) * H_ + 16 * j + mn];
                acc[m][j] = __builtin_amdgcn_wmma_f32_16x16x4_f32(
                    false, a, false, bf, (short)0, acc[m][j], false, false);
            }
        }
    }

    // Store accumulators (C-layout: lane holds rows r+8*hl, col mn)
#pragma unroll
    for (int m = 0; m < 3; ++m) {
#pragma unroll
        for (int j = 0; j < 4; ++j) {
#pragma unroll
            for (int r = 0; r < 8; ++r) {
                const int row = r + 8 * hl;
                outp[m][outBase + (long)row * H_ + 16 * j + mn] = acc[m][j][r];
            }
        }
    }
}

// ---------------------------------------------------------------------------
// Kernel 2: flash-attention forward (causal), one wave per 16-query tile.
// Q fragments cached in VGPRs across the whole key loop; online softmax with
// wave32 intra-half shuffles; P re-laid-out through LDS for the PV WMMA.
// q/k/v (24 MB) fit in the 192 MB L2, so K/V tile re-reads are L2 hits; we
// prefetch the next s-tile's K/V lines (WGP scope) to hide latency behind
// the current tile's 32 WMMAs.
// ---------------------------------------------------------------------------
__global__ __launch_bounds__(32) void flash_attn_kernel(
    const float* __restrict__ q,
    const float* __restrict__ k,
    const float* __restrict__ v,
    float* __restrict__ out)
{
    __shared__ float pLDS[16 * 16];

    const int lane = threadIdx.x;
    const int hl   = lane >> 4;
    const int mn   = lane & 15;

    const int tile = blockIdx.x;                 // over B * (T/16)
    const int b    = tile / (T_ / 16);
    const int t0   = (tile % (T_ / 16)) * 16;
    const long qBase = ((long)b * T_ + t0) * H_;

    // Cache Q A-fragments for the full K-dim (H=64 -> 16 fragments of 4)
    v2f qf[16];
#pragma unroll
    for (int f = 0; f < 16; ++f) {
        const float* p = q + qBase + (long)mn * H_ + 4 * f + 2 * hl;
        qf[f] = *(const v2f*)p;
    }

    v8f o[4] = {};
    float mrow[8], lrow[8];
#pragma unroll
    for (int r = 0; r < 8; ++r) { mrow[r] = -INFINITY; lrow[r] = 0.0f; }

    for (int s0 = 0; s0 <= t0; s0 += 16) {
        const long kBase = ((long)b * T_ + s0) * H_;
        const bool diag  = (s0 == t0);

        // Prefetch next s-tile's K and V rows into the WGP cache (each lane
        // touches one half-row; 32 lanes cover the 16x64 tiles' cachelines).
        if (!diag) {
            const long nBase = kBase + 16 * H_;
            __builtin_prefetch(k + nBase + (long)lane * (H_ / 2), 0, 3);
            __builtin_prefetch(v + nBase + (long)lane * (H_ / 2), 0, 3);
        }

        // ---- S = Q * K^T  (16x16 tile), K-dim = 64 in steps of 4 ----
        v8f s = {};
#pragma unroll
        for (int f = 0; f < 16; ++f) {
            const float* kp = k + kBase + (long)mn * H_ + 4 * f + 2 * hl;
            v2f bf = *(const v2f*)kp;            // B[k][n] = K[s0+n][k]
            s = __builtin_amdgcn_wmma_f32_16x16x4_f32(
                false, qf[f], false, bf, (short)0, s, false, false);
        }

        // ---- scale + causal mask + online softmax (per row r) ----
        float p8[8], corr8[8];
#pragma unroll
        for (int r = 0; r < 8; ++r) {
            float sv = s[r] * SCALE;
            const int m = r + 8 * hl;            // local query row
            if (diag && (mn > m)) sv = -INFINITY;

            // row max across the 16 lanes of this half (wave32 xor shuffles)
            float rm = sv;
            rm = fmaxf(rm, __shfl_xor(rm, 1, 32));
            rm = fmaxf(rm, __shfl_xor(rm, 2, 32));
            rm = fmaxf(rm, __shfl_xor(rm, 4, 32));
            rm = fmaxf(rm, __shfl_xor(rm, 8, 32));

            const float mnew = fmaxf(mrow[r], rm);
            const float corr = expf(mrow[r] - mnew);
            const float pv   = expf(sv - mnew);

            float rs = pv;
            rs += __shfl_xor(rs, 1, 32);
            rs += __shfl_xor(rs, 2, 32);
            rs += __shfl_xor(rs, 4, 32);
            rs += __shfl_xor(rs, 8, 32);

            lrow[r] = lrow[r] * corr + rs;
            mrow[r] = mnew;
            p8[r]   = pv;
            corr8[r] = corr;
        }

        // rescale O accumulators by the correction factor
#pragma unroll
        for (int j = 0; j < 4; ++j)
#pragma unroll
            for (int r = 0; r < 8; ++r)
                o[j][r] *= corr8[r];

        // ---- stage P through LDS: C-layout -> row-major 16x16 ----
        __syncthreads();
#pragma unroll
        for (int r = 0; r < 8; ++r)
            pLDS[(r + 8 * hl) * 16 + mn] = p8[r];
        __syncthreads();

        // ---- O += P * V  (K-dim = 16 in steps of 4, 4 h-tiles) ----
#pragma unroll
        for (int ks = 0; ks < 16; ks += 4) {
            v2f af;
            af[0] = pLDS[mn * 16 + ks + 2 * hl];
            af[1] = pLDS[mn * 16 + ks + 2 * hl + 1];
#pragma unroll
            for (int j = 0; j < 4; ++j) {
                const float* vp = v + kBase + (long)(ks + 2 * hl) * H_ + 16 * j + mn;
                v2f bf;
                bf[0] = vp[0];
                bf[1] = vp[H_];
                o[j] = __builtin_amdgcn_wmma_f32_16x16x4_f32(
                    false, af, false, bf, (short)0, o[j], false, false);
            }
        }
    }

    // ---- normalize and write out ----
#pragma unroll
    for (int j = 0; j < 4; ++j)
#pragma unroll
        for (int r = 0; r < 8; ++r) {
            const int row = r + 8 * hl;
            out[qBase + (long)row * H_ + 16 * j + mn] = o[j][r] / lrow[r];
        }
}

// ---------------------------------------------------------------------------
extern "C" void kernel_launch(void* const* d_in, const int* in_sizes, int n_in,
                              void* d_out, int out_size, void* d_ws, size_t ws_size,
                              hipStream_t stream)
{
    (void)in_sizes; (void)n_in; (void)out_size; (void)ws_size;

    const float* x  = (const float*)d_in[0];
    const float* Wq = (const float*)d_in[1];
    const float* Wk = (const float*)d_in[2];
    const float* Wv = (const float*)d_in[3];
    float* out = (float*)d_out;

    const long BTH = (long)B_ * T_ * H_;   // 2,097,152 floats = 8 MB each
    float* qws = (float*)d_ws;
    float* kws = qws + BTH;
    float* vws = kws + BTH;

    qkv_proj_kernel<<<ROW_TILES, 32, 0, stream>>>(x, Wq, Wk, Wv, qws, kws, vws);
    flash_attn_kernel<<<ROW_TILES, 32, 0, stream>>>(qws, kws, vws, out);
}